// RelativeCoordPredictor_30210799960559
// MI455X (gfx1250) — compile-verified
//
#include <hip/hip_runtime.h>
#include <math.h>

#define N_    64
#define C_    512
#define HW_   784
#define W_    28
#define TILES_ 49   // 784 / 16

typedef float v2f __attribute__((ext_vector_type(2)));
typedef float v8f __attribute__((ext_vector_type(8)));

// ---------------------------------------------------------------------------
// Kernel 1: one block per sample n.
//  - mask[hw] = sum_c x[n,c,hw]   (coalesced: thread t owns hw = t, t+256, ...)
//  - threshold = mean(mask)
//  - argmax of (mask>thr ? mask/512 : 0), first-index tie-break
//  - write basic_anchor, relative_coord_total, stage pw = mask/512 in d_ws
// ---------------------------------------------------------------------------
__global__ __launch_bounds__(256) void rcp_reduce_kernel(
    const float* __restrict__ x, float* __restrict__ out, float* __restrict__ pw)
{
    const int n   = blockIdx.x;
    const int tid = threadIdx.x;
    const float* xn = x + (size_t)n * C_ * HW_;

    const int h0 = tid, h1 = tid + 256, h2 = tid + 512, h3 = tid + 768;
    const bool has3 = (h3 < HW_);

    float s0 = 0.f, s1 = 0.f, s2 = 0.f, s3 = 0.f;
    for (int c = 0; c < C_; ++c) {
        const float* row = xn + (size_t)c * HW_;
        s0 += row[h0];
        s1 += row[h1];
        s2 += row[h2];
        if (has3) s3 += row[h3];
    }

    __shared__ float sh[256];
    __shared__ float av[256];
    __shared__ int   ai[256];
    __shared__ float thr_sh, anch_i, anch_j;

    // ---- threshold = mean over HW ----
    sh[tid] = s0 + s1 + s2 + s3;
    __syncthreads();
    for (int ofs = 128; ofs > 0; ofs >>= 1) {
        if (tid < ofs) sh[tid] += sh[tid + ofs];
        __syncthreads();
    }
    if (tid == 0) thr_sh = sh[0] / (float)HW_;
    __syncthreads();
    const float thr  = thr_sh;
    const float invC = 1.0f / (float)C_;

    // ---- per-thread argmax over owned positions (increasing index order) ----
    float bv = -INFINITY; int bi = 0;
    {
        float sc;
        sc = (s0 > thr) ? s0 * invC : 0.0f; if (sc > bv) { bv = sc; bi = h0; }
        sc = (s1 > thr) ? s1 * invC : 0.0f; if (sc > bv) { bv = sc; bi = h1; }
        sc = (s2 > thr) ? s2 * invC : 0.0f; if (sc > bv) { bv = sc; bi = h2; }
        if (has3) {
            sc = (s3 > thr) ? s3 * invC : 0.0f; if (sc > bv) { bv = sc; bi = h3; }
        }
    }
    av[tid] = bv; ai[tid] = bi;
    __syncthreads();
    for (int ofs = 128; ofs > 0; ofs >>= 1) {
        if (tid < ofs) {
            float ov = av[tid + ofs]; int oi = ai[tid + ofs];
            if (ov > av[tid] || (ov == av[tid] && oi < ai[tid])) {
                av[tid] = ov; ai[tid] = oi;
            }
        }
        __syncthreads();
    }
    if (tid == 0) {
        int idx = ai[0];
        float fi = (float)(idx / W_), fj = (float)(idx % W_);
        anch_i = fi; anch_j = fj;
        out[(size_t)N_ * HW_ * 2 + n * 2 + 0] = fi;   // basic_anchor
        out[(size_t)N_ * HW_ * 2 + n * 2 + 1] = fj;
    }
    __syncthreads();
    const float aiF = anch_i, ajF = anch_j;
    const float PI_F = 3.14159265358979323846f;

    // ---- relative coords + stage pw ----
    float* rel = out + (size_t)n * HW_ * 2;
    float* pwn = pw + (size_t)n * HW_;
    float msk[4] = { s0, s1, s2, s3 };
    int   hws[4] = { h0, h1, h2, h3 };
    #pragma unroll
    for (int k = 0; k < 4; ++k) {
        int hw = hws[k];
        if (hw < HW_) {
            pwn[hw] = msk[k] * invC;
            float ri = ((float)(hw / W_) - aiF) / (float)W_;
            float rj = ((float)(hw % W_) - ajF) / (float)W_;
            float dist = sqrtf(ri * ri + rj * rj);
            float ang  = (atan2f(rj, ri) / PI_F + 1.0f) * 0.5f;
            rel[hw * 2 + 0] = dist;
            rel[hw * 2 + 1] = ang;
        }
    }
}

// ---------------------------------------------------------------------------
// Kernel 2: rank-1 outer product via V_WMMA_F32_16X16X4_F32.
// One wave per (n, tm) row strip; A holds u in the K=0 column (VGPR0 lanes
// 0-15), B holds v in the K=0 row; all other K slices zero -> D = u_m * v_n
// exactly (f32, RNE). Store-bound: 157 MB of position_weight.
// ---------------------------------------------------------------------------
__global__ __launch_bounds__(256) void rcp_outer_wmma_kernel(
    const float* __restrict__ pw, float* __restrict__ pos)
{
    const int wave = (int)((blockIdx.x * 256u + threadIdx.x) >> 5);
    const int lane = threadIdx.x & 31;
    const int nn = wave / TILES_;
    const int tm = wave % TILES_;
    if (nn >= N_) return;   // wave-uniform; grid is exact so never taken

    const float* p   = pw + (size_t)nn * HW_;
    const int    l15 = lane & 15;

    // A-matrix 16x4: VGPR0 lanes0-15 = K=0 (M=lane); everything else zero.
    float uload = p[tm * 16 + l15];
    float uval  = (lane < 16) ? uload : 0.0f;
    v2f a; a.x = uval; a.y = 0.0f;

    // D layout: VGPR r -> M = r (+8 for lanes 16-31), N = lane&15
    float* base = pos + (size_t)nn * HW_ * HW_
                      + (size_t)(tm * 16 + ((lane < 16) ? 0 : 8)) * HW_
                      + l15;

    for (int tn = 0; tn < TILES_; ++tn) {
        float vload = p[tn * 16 + l15];
        float vval  = (lane < 16) ? vload : 0.0f;
        v2f b; b.x = vval; b.y = 0.0f;

        v8f c = {0.f, 0.f, 0.f, 0.f, 0.f, 0.f, 0.f, 0.f};
        v8f d = __builtin_amdgcn_wmma_f32_16x16x4_f32(
            /*neg_a=*/false, a, /*neg_b=*/false, b,
            /*c_mod=*/(short)0, c, /*reuse_a=*/false, /*reuse_b=*/false);

        float* o = base + tn * 16;
        #pragma unroll
        for (int r = 0; r < 8; ++r)
            o[(size_t)r * HW_] = d[r];
    }
}

extern "C" void kernel_launch(void* const* d_in, const int* in_sizes, int n_in,
                              void* d_out, int out_size, void* d_ws, size_t ws_size,
                              hipStream_t stream) {
    const float* x  = (const float*)d_in[0];
    float* out      = (float*)d_out;
    float* pw       = (float*)d_ws;                       // 64*784 floats staged
    float* pos      = out + (size_t)N_ * HW_ * 2 + N_ * 2; // position_weight

    rcp_reduce_kernel<<<N_, 256, 0, stream>>>(x, out, pw);

    const int waves  = N_ * TILES_;        // 3136 waves
    const int blocks = (waves * 32) / 256; // 392 blocks of 8 waves
    rcp_outer_wmma_kernel<<<blocks, 256, 0, stream>>>(pw, pos);
}